// MambaEncoder_12292196402034
// MI455X (gfx1250) — compile-verified
//
#include <hip/hip_runtime.h>
#include <math.h>

// ---------------------------------------------------------------------------
// MambaEncoder on MI455X (gfx1250, wave32).
//  - All matmuls via V_WMMA_F32_16X16X4_F32 (fp32-exact vs reference).
//  - Big GEMMs: 2x2 WMMA tiles per wave (32x32 out) -> each A/B fragment
//    feeds 2 wmma ops, halving VMEM bytes per flop vs 1x1 tiling.
//  - Selective scan: 1 block per sequence, 1 thread per channel, 16-wide SSM
//    state in registers, sequential over T (latency-bound by construction).
//  - Working set (~90MB) is L2-resident (192MB); HBM traffic is negligible.
// ---------------------------------------------------------------------------

typedef __attribute__((ext_vector_type(2))) float v2f;
typedef __attribute__((ext_vector_type(4))) float v4f;
typedef __attribute__((ext_vector_type(8))) float v8f;

#define NSEQ 64
#define TT   256
#define FF   64
#define DMm  128
#define DIi  256
#define DSs  16
#define DTRr 8
#define DCc  4
#define XDBL 40            // DTR + 2*DS
#define MROWS (NSEQ * TT)  // 16384

#define EPI_NONE 0
#define EPI_BIAS 1
#define EPI_BIAS_SOFTPLUS 2

#if defined(__has_builtin)
#if __has_builtin(__builtin_amdgcn_wmma_f32_16x16x4_f32)
#define HAVE_WMMA_F32 1
#endif
#endif

__device__ __forceinline__ v8f wmma_f32(v2f a, v2f b, v8f c) {
#ifdef HAVE_WMMA_F32
  return __builtin_amdgcn_wmma_f32_16x16x4_f32(
      /*neg_a=*/false, a, /*neg_b=*/false, b,
      /*c_mod=*/(short)0, c, /*reuse_a=*/false, /*reuse_b=*/false);
#else
  return c;  // compile-safety stub only
#endif
}

// x: [N, F, T]  ->  xT: [N*T, F]
__global__ void k_transpose_x(const float* __restrict__ x, float* __restrict__ xT) {
  int idx = blockIdx.x * blockDim.x + threadIdx.x;   // over N*T*F
  int f = idx & (FF - 1);
  int t = (idx >> 6) & (TT - 1);
  int n = idx >> 14;
  xT[idx] = x[((size_t)n * FF + f) * TT + t];
}

// -------------------- 2x2-tile WMMA GEMM (M%32==0, Nout%32==0) -------------
// out[m, n] = sum_k A[m, k] * W[n, k]  (+ epilogue). One 32x32 tile per wave:
// 4 wmma per 4 float2 loads. Fragment layouts per CDNA5 ISA 7.12.2.
template <int EPI>
__global__ void __launch_bounds__(256) k_gemm_wmma_2x2(
    const float* __restrict__ A, int lda,
    const float* __restrict__ W,      // [Nout, K] row-major (weight)
    const float* __restrict__ bias,   // [Nout] or nullptr
    float* __restrict__ out, int ldo,
    int M, int Nout, int K)
{
  const int lane = threadIdx.x & 31;
  const int ntn  = Nout >> 5;
  const int wid  = blockIdx.x * (blockDim.x >> 5) + (threadIdx.x >> 5);
  const int ntiles = (M >> 5) * ntn;
  if (wid >= ntiles) return;                 // uniform per wave: EXEC all-1s

  const int tn = wid % ntn;
  const int tm = wid / ntn;
  const int m0 = tm << 5;
  const int n0 = tn << 5;
  const int half = lane >> 4;                // 0: K pair {0,1}; 1: K pair {2,3}
  const int lm   = lane & 15;

  const float* Ap0 = A + (size_t)(m0 + lm) * lda + 2 * half;
  const float* Ap1 = Ap0 + (size_t)16 * lda;
  const float* Wp0 = W + (size_t)(n0 + lm) * K + 2 * half;
  const float* Wp1 = Wp0 + (size_t)16 * K;

  v8f c00 = {}, c01 = {}, c10 = {}, c11 = {};
  for (int k0 = 0; k0 < K; k0 += 4) {
    if ((k0 & 63) == 0) {                    // gfx1250 global_prefetch_b8
      __builtin_prefetch(Ap0 + k0 + 64, 0, 3);
      __builtin_prefetch(Ap1 + k0 + 64, 0, 3);
      __builtin_prefetch(Wp0 + k0 + 64, 0, 3);
      __builtin_prefetch(Wp1 + k0 + 64, 0, 3);
    }
    v2f a0 = *(const v2f*)(Ap0 + k0);
    v2f a1 = *(const v2f*)(Ap1 + k0);
    v2f b0 = *(const v2f*)(Wp0 + k0);
    v2f b1 = *(const v2f*)(Wp1 + k0);
    c00 = wmma_f32(a0, b0, c00);
    c01 = wmma_f32(a0, b1, c01);
    c10 = wmma_f32(a1, b0, c10);
    c11 = wmma_f32(a1, b1, c11);
  }

  float bv0 = 0.f, bv1 = 0.f;
  if (EPI != EPI_NONE) { bv0 = bias[n0 + lm]; bv1 = bias[n0 + 16 + lm]; }

  // c[r] -> row (mbase + r + 8*half), fixed column
  #pragma unroll
  for (int r = 0; r < 8; ++r) {
    const int mr = r + 8 * half;
    float v00 = c00[r] + bv0, v01 = c01[r] + bv1;
    float v10 = c10[r] + bv0, v11 = c11[r] + bv1;
    if (EPI == EPI_BIAS_SOFTPLUS) {
      v00 = (v00 > 20.f) ? v00 : log1pf(__expf(v00));
      v01 = (v01 > 20.f) ? v01 : log1pf(__expf(v01));
      v10 = (v10 > 20.f) ? v10 : log1pf(__expf(v10));
      v11 = (v11 > 20.f) ? v11 : log1pf(__expf(v11));
    }
    out[(size_t)(m0 + mr) * ldo + n0 + lm]            = v00;
    out[(size_t)(m0 + mr) * ldo + n0 + 16 + lm]       = v01;
    out[(size_t)(m0 + 16 + mr) * ldo + n0 + lm]       = v10;
    out[(size_t)(m0 + 16 + mr) * ldo + n0 + 16 + lm]  = v11;
  }
}

// -------------------- 1x1-tile GEMM for ragged Nout (x_proj, Nout=40) ------
__global__ void __launch_bounds__(256) k_gemm_wmma_ragged(
    const float* __restrict__ A, int lda,
    const float* __restrict__ W, float* __restrict__ out, int ldo,
    int M, int Nout, int K)
{
  const int lane = threadIdx.x & 31;
  const int ntn  = (Nout + 15) >> 4;
  const int wid  = blockIdx.x * (blockDim.x >> 5) + (threadIdx.x >> 5);
  const int ntiles = (M >> 4) * ntn;
  if (wid >= ntiles) return;

  const int tn = wid % ntn;
  const int tm = wid / ntn;
  const int m0 = tm << 4;
  const int n0 = tn << 4;
  const int half = lane >> 4;
  const int lm   = lane & 15;

  const int  wrow = n0 + lm;
  const bool wok  = (wrow < Nout);
  const float* Ap = A + (size_t)(m0 + lm) * lda + 2 * half;
  const float* Wp = W + (size_t)(wok ? wrow : 0) * K + 2 * half;

  v8f c = {};
  for (int k0 = 0; k0 < K; k0 += 4) {
    v2f a = *(const v2f*)(Ap + k0);
    v2f b = *(const v2f*)(Wp + k0);
    if (!wok) { b.x = 0.f; b.y = 0.f; }
    c = wmma_f32(a, b, c);
  }
  #pragma unroll
  for (int r = 0; r < 8; ++r)
    if (wok) out[(size_t)(m0 + r + 8 * half) * ldo + wrow] = c[r];
}

// Causal depthwise conv (k=4) + bias + SiLU:  u[n,t,d]
__global__ void k_conv_silu(const float* __restrict__ xz,
                            const float* __restrict__ cw,   // [DI, 4]
                            const float* __restrict__ cb,   // [DI]
                            float* __restrict__ u)
{
  int idx = blockIdx.x * blockDim.x + threadIdx.x;   // N*T*DI
  int d = idx & (DIi - 1);
  int t = (idx >> 8) & (TT - 1);
  int n = idx >> 16;
  v4f w = *(const v4f*)(cw + d * DCc);
  float acc = cb[d];
  #pragma unroll
  for (int j = 0; j < DCc; ++j) {
    int ts = t - (DCc - 1) + j;
    float xv = (ts >= 0) ? xz[((size_t)n * TT + ts) * (2 * DIi) + d] : 0.f;
    acc = fmaf(xv, w[j], acc);
  }
  u[idx] = acc / (1.f + __expf(-acc));               // silu
}

// Selective scan, fused with u*D residual, silu(z) gate. Writes over u.
__global__ void __launch_bounds__(256) k_scan(
    const float* __restrict__ delta,   // [N*T, DI]
    const float* __restrict__ xdbl,    // [N*T, 40]  (dt | B | C)
    const float* __restrict__ xz,      // [N*T, 512] (z at cols 256..511)
    const float* __restrict__ alog,    // [DI, DS]  (this layer)
    const float* __restrict__ Dv,      // [DI]
    float* __restrict__ u)             // in: u ; out: gated y
{
  const int n = blockIdx.x;
  const int d = threadIdx.x;

  float Ar[DSs];
  #pragma unroll
  for (int s = 0; s < DSs; ++s) Ar[s] = -__expf(alog[d * DSs + s]);
  const float Dd = Dv[d];

  float hst[DSs];
  #pragma unroll
  for (int s = 0; s < DSs; ++s) hst[s] = 0.f;

  for (int t = 0; t < TT; ++t) {
    const size_t row = (size_t)n * TT + t;
    float dlt = delta[row * DIi + d];
    float ut  = u[row * DIi + d];
    float zt  = xz[row * (2 * DIi) + DIi + d];
    const float* bc = xdbl + row * XDBL + DTRr;      // 16B-aligned
    v4f Bv[4], Cv[4];
    #pragma unroll
    for (int q = 0; q < 4; ++q) Bv[q] = *(const v4f*)(bc + 4 * q);
    #pragma unroll
    for (int q = 0; q < 4; ++q) Cv[q] = *(const v4f*)(bc + 16 + 4 * q);

    float du = dlt * ut;
    float y = 0.f;
    #pragma unroll
    for (int s = 0; s < DSs; ++s) {
      float dA = __expf(dlt * Ar[s]);
      hst[s] = fmaf(dA, hst[s], du * Bv[s >> 2][s & 3]);
      y = fmaf(hst[s], Cv[s >> 2][s & 3], y);
    }
    y = fmaf(ut, Dd, y);
    float sz = zt / (1.f + __expf(-zt));
    u[row * DIi + d] = y * sz;
  }
}

// hout = LayerNorm(hin + yin) * w + b ; one wave32 per 128-wide row.
__global__ void __launch_bounds__(256) k_layernorm(
    const float* __restrict__ hin, const float* __restrict__ yin,
    const float* __restrict__ w, const float* __restrict__ b,
    float* __restrict__ hout, int M)
{
  const int lane = threadIdx.x & 31;
  const int row  = blockIdx.x * 8 + (threadIdx.x >> 5);
  if (row >= M) return;
  const size_t base = (size_t)row * DMm;
  v4f x = *(const v4f*)(hin + base + lane * 4);
  if (yin) {
    v4f y = *(const v4f*)(yin + base + lane * 4);
    x += y;
  }
  float s  = x[0] + x[1] + x[2] + x[3];
  float sq = x[0]*x[0] + x[1]*x[1] + x[2]*x[2] + x[3]*x[3];
  #pragma unroll
  for (int m = 16; m >= 1; m >>= 1) {
    s  += __shfl_xor(s,  m, 32);
    sq += __shfl_xor(sq, m, 32);
  }
  float mu  = s * (1.f / DMm);
  float var = sq * (1.f / DMm) - mu * mu;
  float rs  = rsqrtf(var + 1e-5f);
  v4f wv = *(const v4f*)(w + lane * 4);
  v4f bv = *(const v4f*)(b + lane * 4);
  v4f o;
  #pragma unroll
  for (int i = 0; i < 4; ++i) o[i] = (x[i] - mu) * rs * wv[i] + bv[i];
  *(v4f*)(hout + base + lane * 4) = o;
}

// out[n, d] = mean_t h[n, t, d]
__global__ void k_mean_t(const float* __restrict__ h, float* __restrict__ out)
{
  int idx = blockIdx.x * blockDim.x + threadIdx.x;   // N*DM = 8192
  int d = idx & (DMm - 1);
  int n = idx >> 7;
  float acc = 0.f;
  for (int t = 0; t < TT; ++t) acc += h[((size_t)n * TT + t) * DMm + d];
  out[idx] = acc * (1.f / TT);
}

extern "C" void kernel_launch(void* const* d_in, const int* in_sizes, int n_in,
                              void* d_out, int out_size, void* d_ws, size_t ws_size,
                              hipStream_t stream) {
  const float* x     = (const float*)d_in[0];
  const float* inp_w = (const float*)d_in[1];
  const float* inp_b = (const float*)d_in[2];
  const float* ipw   = (const float*)d_in[3];   // [2, 512, 128]
  const float* cw    = (const float*)d_in[4];   // [2, 256, 4]
  const float* cb    = (const float*)d_in[5];   // [2, 256]
  const float* xpw   = (const float*)d_in[6];   // [2, 40, 256]
  const float* dtw   = (const float*)d_in[7];   // [2, 256, 8]
  const float* dtb   = (const float*)d_in[8];   // [2, 256]
  const float* alog  = (const float*)d_in[9];   // [2, 256, 16]
  const float* Dv    = (const float*)d_in[10];  // [2, 256]
  const float* opw   = (const float*)d_in[11];  // [2, 128, 256]
  const float* nw    = (const float*)d_in[12];  // [2, 128]
  const float* nb    = (const float*)d_in[13];  // [2, 128]
  const float* onw   = (const float*)d_in[14];  // [128]
  const float* onb   = (const float*)d_in[15];  // [128]
  float* out = (float*)d_out;

  float* ws = (float*)d_ws;
  size_t o = 0;
  float* xT   = ws + o; o += (size_t)MROWS * FF;        // [16384, 64]
  float* h    = ws + o; o += (size_t)MROWS * DMm;       // [16384, 128]
  float* xz   = ws + o; o += (size_t)MROWS * 2 * DIi;   // [16384, 512]
  float* u    = ws + o; o += (size_t)MROWS * DIi;       // [16384, 256]
  float* xdbl = ws + o; o += (size_t)MROWS * XDBL;      // [16384, 40]
  float* dlt  = ws + o; o += (size_t)MROWS * DIi;       // [16384, 256]
  float* ypj  = ws + o;                                 // [16384, 128]

  auto launch_2x2 = [&](int epi, const float* A, int lda, const float* W,
                        const float* bias, float* O, int ldo,
                        int M, int Nn, int K) {
    int ntiles = (M / 32) * (Nn / 32);
    int blocks = (ntiles + 7) / 8;                      // 8 waves / 256-thread block
    switch (epi) {
      case EPI_BIAS:
        k_gemm_wmma_2x2<EPI_BIAS><<<blocks, 256, 0, stream>>>(
            A, lda, W, bias, O, ldo, M, Nn, K);
        break;
      case EPI_BIAS_SOFTPLUS:
        k_gemm_wmma_2x2<EPI_BIAS_SOFTPLUS><<<blocks, 256, 0, stream>>>(
            A, lda, W, bias, O, ldo, M, Nn, K);
        break;
      default:
        k_gemm_wmma_2x2<EPI_NONE><<<blocks, 256, 0, stream>>>(
            A, lda, W, nullptr, O, ldo, M, Nn, K);
        break;
    }
  };

  // h0 = xT @ inp_w.T + inp_b
  k_transpose_x<<<(MROWS * FF) / 256, 256, 0, stream>>>(x, xT);
  launch_2x2(EPI_BIAS, xT, FF, inp_w, inp_b, h, DMm, MROWS, DMm, FF);

  for (int l = 0; l < 2; ++l) {
    // xz = h @ in_proj_w.T  -> [.., 512] (xi = cols 0..255, z = cols 256..511)
    launch_2x2(EPI_NONE, h, DMm, ipw + (size_t)l * 2 * DIi * DMm, nullptr,
               xz, 2 * DIi, MROWS, 2 * DIi, DMm);
    // u = silu(conv1d(xi) + cb)
    k_conv_silu<<<(MROWS * DIi) / 256, 256, 0, stream>>>(
        xz, cw + (size_t)l * DIi * DCc, cb + (size_t)l * DIi, u);
    // x_dbl = u @ x_proj_w.T   (Nout=40: ragged 1x1-tile path)
    {
      int ntiles = (MROWS / 16) * ((XDBL + 15) / 16);
      int blocks = (ntiles + 7) / 8;
      k_gemm_wmma_ragged<<<blocks, 256, 0, stream>>>(
          u, DIi, xpw + (size_t)l * XDBL * DIi, xdbl, XDBL, MROWS, XDBL, DIi);
    }
    // delta = softplus(dt @ dt_w.T + dt_b)   (A = x_dbl cols 0..7, lda=40)
    launch_2x2(EPI_BIAS_SOFTPLUS, xdbl, XDBL, dtw + (size_t)l * DIi * DTRr,
               dtb + (size_t)l * DIi, dlt, DIi, MROWS, DIi, DTRr);
    // scan + u*D + silu(z) gate, written in place over u
    k_scan<<<NSEQ, DIi, 0, stream>>>(
        dlt, xdbl, xz, alog + (size_t)l * DIi * DSs, Dv + (size_t)l * DIi, u);
    // ypj = y @ out_proj_w.T
    launch_2x2(EPI_NONE, u, DIi, opw + (size_t)l * DMm * DIi, nullptr,
               ypj, DMm, MROWS, DMm, DIi);
    // h = LN(h + ypj)
    k_layernorm<<<MROWS / 8, 256, 0, stream>>>(
        h, ypj, nw + (size_t)l * DMm, nb + (size_t)l * DMm, h, MROWS);
  }

  // h = LN(h); out = mean over T
  k_layernorm<<<MROWS / 8, 256, 0, stream>>>(h, nullptr, onw, onb, h, MROWS);
  k_mean_t<<<(NSEQ * DMm) / 256, 256, 0, stream>>>(h, out);
}